// GCNEncoder_19301583028533
// MI455X (gfx1250) — compile-verified
//
#include <hip/hip_runtime.h>
#include <hip/hip_bf16.h>

#define N_NODES 50000
#define N_EDGES 1600000
#define IN_DIM  128
#define HID     128
#define OUT_DIM 64
#define BN_EPS  1e-5f

typedef float v2f __attribute__((ext_vector_type(2)));
typedef float v8f __attribute__((ext_vector_type(8)));

// ---------------------------------------------------------------- degrees ---
__global__ void gcn_init_deg(float* __restrict__ deg, int n) {
    int i = blockIdx.x * blockDim.x + threadIdx.x;
    if (i < n) deg[i] = 1.0f;               // self-loop contributes 1
}

__global__ void gcn_deg_accum(const int* __restrict__ dst, float* __restrict__ deg, int e) {
    int i = blockIdx.x * blockDim.x + threadIdx.x;
    if (i < e) atomicAdd(&deg[dst[i]], 1.0f);
}

__global__ void gcn_dinv(float* __restrict__ deg, int n) {
    int i = blockIdx.x * blockDim.x + threadIdx.x;
    if (i < n) {
        float d = deg[i];
        deg[i] = (d > 0.0f) ? rsqrtf(d) : 0.0f;   // in-place: deg -> dinv
    }
}

// ------------------------------------------------------- WMMA f32 GEMM ------
// C[M,NDIM] = A[M,KDIM] @ B[KDIM,NDIM]   (one wave per 16x16 output tile)
template<int KDIM, int NDIM>
__global__ __launch_bounds__(256) void gcn_gemm_wmma_f32(
    const float* __restrict__ A, const float* __restrict__ B,
    float* __restrict__ C, int M)
{
    constexpr int TILES_N = NDIM / 16;
    const int lane  = threadIdx.x & 31;
    const int wave  = blockIdx.x * (blockDim.x >> 5) + (threadIdx.x >> 5);
    const int tileM = wave / TILES_N;
    const int tileN = wave - tileM * TILES_N;
    if (tileM * 16 >= M) return;            // wave-uniform: EXEC stays all-1s

    const int lm = lane & 15;               // row (A) / col (B,C) within tile
    const int kh = lane >> 4;               // K-half select per ISA layout

    const float* arow = A + (size_t)(tileM * 16 + lm) * KDIM + 2 * kh;
    const float* bcol = B + (size_t)(2 * kh) * NDIM + tileN * 16 + lm;

    v8f acc = {0.f, 0.f, 0.f, 0.f, 0.f, 0.f, 0.f, 0.f};
#pragma unroll
    for (int kb = 0; kb < KDIM; kb += 4) {
        v2f a, b;
        a.x = arow[kb + 0];                         // A[m][kb + 2*kh + 0]
        a.y = arow[kb + 1];                         // A[m][kb + 2*kh + 1]
        b.x = bcol[(size_t)(kb + 0) * NDIM];        // B[kb + 2*kh + 0][n]
        b.y = bcol[(size_t)(kb + 1) * NDIM];        // B[kb + 2*kh + 1][n]
        acc = __builtin_amdgcn_wmma_f32_16x16x4_f32(
            /*neg_a=*/false, a, /*neg_b=*/false, b,
            /*c_mod=*/(short)0, acc, /*reuse_a=*/false, /*reuse_b=*/false);
    }

    // C/D layout: VGPR r -> M = r (lanes 0-15) / r+8 (lanes 16-31), N = lane&15
    float* crow = C + (size_t)(tileM * 16 + 8 * kh) * NDIM + tileN * 16 + lm;
#pragma unroll
    for (int r = 0; r < 8; ++r)
        crow[(size_t)r * NDIM] = acc[r];
}

// ------------------------------------------------ aggregation (scatter) -----
// out[n,:] = dinv[n]^2 * h[n,:] + bias   (self-loop term + post-agg bias)
template<int F>
__global__ void gcn_agg_init(const float* __restrict__ h,
                             const float* __restrict__ dinv,
                             const float* __restrict__ bias,
                             float* __restrict__ out, int n)
{
    int i = blockIdx.x * blockDim.x + threadIdx.x;
    if (i >= n * F) return;
    int node = i / F;
    int f = i - node * F;
    float di = dinv[node];
    out[i] = di * di * h[i] + bias[f];
}

// one thread per (edge, 4-feature chunk); lanes within a group share the edge
template<int F>
__global__ void gcn_edge_scatter(const int* __restrict__ src,
                                 const int* __restrict__ dst,
                                 const float* __restrict__ dinv,
                                 const float* __restrict__ h,
                                 float* __restrict__ out, int e)
{
    constexpr int CH = F / 4;
    int i  = blockIdx.x * blockDim.x + threadIdx.x;
    int ed = i / CH;
    if (ed >= e) return;
    int c = (i - ed * CH) * 4;
    int s = src[ed];
    int d = dst[ed];
    float nrm = dinv[s] * dinv[d];
    float4 v = *(const float4*)(h + (size_t)s * F + c);
    float* o = out + (size_t)d * F + c;
    atomicAdd(o + 0, nrm * v.x);
    atomicAdd(o + 1, nrm * v.y);
    atomicAdd(o + 2, nrm * v.z);
    atomicAdd(o + 3, nrm * v.w);
}

// -------------------------------------------------------- BatchNorm+PReLU ---
__global__ void gcn_zero(float* __restrict__ p, int n) {
    int i = blockIdx.x * blockDim.x + threadIdx.x;
    if (i < n) p[i] = 0.0f;
}

__global__ void gcn_bn_stats(const float* __restrict__ g, float* __restrict__ stats, int n) {
    int f = threadIdx.x;                    // 128 threads: one feature each
    float s = 0.0f, s2 = 0.0f;
    for (int r = blockIdx.x; r < n; r += gridDim.x) {
        float v = g[(size_t)r * HID + f];
        s  += v;
        s2 += v * v;
    }
    atomicAdd(&stats[f],        s);
    atomicAdd(&stats[HID + f], s2);
}

__global__ void gcn_bn_finalize(const float* __restrict__ stats,
                                const float* __restrict__ gamma,
                                const float* __restrict__ beta,
                                float* __restrict__ ss, int n)
{
    int f = threadIdx.x;                    // 128 threads
    float inv_n = 1.0f / (float)n;
    float mu  = stats[f] * inv_n;
    float var = stats[HID + f] * inv_n - mu * mu;
    float rs  = rsqrtf(var + BN_EPS);
    float sc  = gamma[f] * rs;
    ss[f]        = sc;
    ss[HID + f]  = beta[f] - mu * sc;
}

__global__ void gcn_bn_prelu(float* __restrict__ g, const float* __restrict__ ss,
                             const float* __restrict__ prelu_a, int total)
{
    int i = blockIdx.x * blockDim.x + threadIdx.x;
    if (i >= total) return;
    int f = i & (HID - 1);
    float v = g[i] * ss[f] + ss[HID + f];
    float a = prelu_a[0];
    g[i] = (v > 0.0f) ? v : a * v;
}

// -------------------------------------------------------------- launcher ----
extern "C" void kernel_launch(void* const* d_in, const int* in_sizes, int n_in,
                              void* d_out, int out_size, void* d_ws, size_t ws_size,
                              hipStream_t stream)
{
    const float* x     = (const float*)d_in[0];
    const int*   ei    = (const int*)  d_in[1];     // [2, E] flat: row0=src row1=dst
    const float* W1    = (const float*)d_in[2];
    const float* b1    = (const float*)d_in[3];
    const float* W2    = (const float*)d_in[4];
    const float* b2    = (const float*)d_in[5];
    const float* gamma = (const float*)d_in[6];
    const float* beta  = (const float*)d_in[7];
    const float* pa    = (const float*)d_in[8];
    float* out = (float*)d_out;

    const int* src = ei;
    const int* dst = ei + N_EDGES;

    // workspace layout (256B-aligned slabs)
    char* ws = (char*)d_ws;
    size_t off = 0;
    auto take = [&](size_t bytes) { char* p = ws + off; off += (bytes + 255) & ~(size_t)255; return p; };
    float* dinv  = (float*)take((size_t)N_NODES * sizeof(float));
    float* h1    = (float*)take((size_t)N_NODES * HID * sizeof(float));
    float* g1    = (float*)take((size_t)N_NODES * HID * sizeof(float));
    float* h2    = (float*)take((size_t)N_NODES * OUT_DIM * sizeof(float));
    float* stats = (float*)take(2 * HID * sizeof(float));
    float* ss    = (float*)take(2 * HID * sizeof(float));
    (void)ws_size; (void)in_sizes; (void)n_in; (void)out_size;

    const int B = 256;

    // ---- normalization weights
    gcn_init_deg <<<(N_NODES + B - 1) / B, B, 0, stream>>>(dinv, N_NODES);
    gcn_deg_accum<<<(N_EDGES + B - 1) / B, B, 0, stream>>>(dst, dinv, N_EDGES);
    gcn_dinv     <<<(N_NODES + B - 1) / B, B, 0, stream>>>(dinv, N_NODES);

    // ---- layer 1: H1 = X @ W1 (WMMA f32)
    {
        int tilesM = (N_NODES + 15) / 16;           // 3125
        int waves  = tilesM * (HID / 16);           // 25000
        gcn_gemm_wmma_f32<IN_DIM, HID><<<(waves + 7) / 8, B, 0, stream>>>(x, W1, h1, N_NODES);
    }
    // G1 = dinv^2*H1 + b1, then scatter-add edge messages
    gcn_agg_init<HID><<<((size_t)N_NODES * HID + B - 1) / B, B, 0, stream>>>(h1, dinv, b1, g1, N_NODES);
    gcn_edge_scatter<HID><<<((size_t)N_EDGES * (HID / 4) + B - 1) / B, B, 0, stream>>>(
        src, dst, dinv, h1, g1, N_EDGES);

    // ---- BatchNorm (batch stats) + PReLU, in place on G1
    gcn_zero<<<1, 2 * HID, 0, stream>>>(stats, 2 * HID);
    gcn_bn_stats   <<<256, HID, 0, stream>>>(g1, stats, N_NODES);
    gcn_bn_finalize<<<1,   HID, 0, stream>>>(stats, gamma, beta, ss, N_NODES);
    gcn_bn_prelu<<<((size_t)N_NODES * HID + B - 1) / B, B, 0, stream>>>(
        g1, ss, pa, N_NODES * HID);

    // ---- layer 2: H2 = G1 @ W2 (WMMA f32)
    {
        int tilesM = (N_NODES + 15) / 16;
        int waves  = tilesM * (OUT_DIM / 16);       // 12500
        gcn_gemm_wmma_f32<HID, OUT_DIM><<<(waves + 7) / 8, B, 0, stream>>>(g1, W2, h2, N_NODES);
    }
    // out = dinv^2*H2 + b2, then scatter-add edge messages
    gcn_agg_init<OUT_DIM><<<((size_t)N_NODES * OUT_DIM + B - 1) / B, B, 0, stream>>>(
        h2, dinv, b2, out, N_NODES);
    gcn_edge_scatter<OUT_DIM><<<((size_t)N_EDGES * (OUT_DIM / 4) + B - 1) / B, B, 0, stream>>>(
        src, dst, dinv, h2, out, N_EDGES);
}